// Adversarial_9045201125868
// MI455X (gfx1250) — compile-verified
//
#include <hip/hip_runtime.h>
#include <hip/hip_bf16.h>

typedef float v2f __attribute__((ext_vector_type(2)));
typedef float v4f __attribute__((ext_vector_type(4)));
typedef float v8f __attribute__((ext_vector_type(8)));

#define TLEN 2048
#define BZ   32
#define EMB  1024
#define EPSILON 0.05f

// ---------------------------------------------------------------------------
// Threefry-2x32-20 (JAX PRNG), key = (0, 42) for jax.random.key(42)
// ---------------------------------------------------------------------------
__device__ __forceinline__ void threefry2x32(unsigned k0, unsigned k1,
                                             unsigned c0, unsigned c1,
                                             unsigned& o0, unsigned& o1) {
  unsigned ks2 = k0 ^ k1 ^ 0x1BD11BDAu;
  unsigned x0 = c0 + k0, x1 = c1 + k1;
#define TF_ROUND(r) { x0 += x1; x1 = (x1 << (r)) | (x1 >> (32 - (r))); x1 ^= x0; }
  TF_ROUND(13) TF_ROUND(15) TF_ROUND(26) TF_ROUND(6)
  x0 += k1;  x1 += ks2 + 1u;
  TF_ROUND(17) TF_ROUND(29) TF_ROUND(16) TF_ROUND(24)
  x0 += ks2; x1 += k0 + 2u;
  TF_ROUND(13) TF_ROUND(15) TF_ROUND(26) TF_ROUND(6)
  x0 += k0;  x1 += k1 + 3u;
  TF_ROUND(17) TF_ROUND(29) TF_ROUND(16) TF_ROUND(24)
  x0 += k1;  x1 += ks2 + 4u;
  TF_ROUND(13) TF_ROUND(15) TF_ROUND(26) TF_ROUND(6)
  x0 += ks2; x1 += k0 + 5u;
#undef TF_ROUND
  o0 = x0; o1 = x1;
}

// ---------------------------------------------------------------------------
// K1: one wave32 per timestep t. Masked Gumbel argmax over 32 batch slots.
// ---------------------------------------------------------------------------
__global__ __launch_bounds__(32) void select_kernel(const int* __restrict__ dpad,
                                                    int* __restrict__ Lws) {
  const int t    = blockIdx.x;
  const int lane = threadIdx.x;                 // 0..31 (wave32)
  const unsigned idx  = (unsigned)(t * BZ + lane);
  const unsigned half = (TLEN * BZ) / 2;        // JAX counter split point
  unsigned c0, c1, o0, o1;
  if (idx < half) { c0 = idx;        c1 = idx + half; }
  else            { c0 = idx - half; c1 = idx;        }
  threefry2x32(0u, 42u, c0, c1, o0, o1);
  const unsigned bits = (idx < half) ? o0 : o1;

  // JAX bits->uniform in [tiny, 1), then Gumbel
  float u = __uint_as_float((bits >> 9) | 0x3f800000u) - 1.0f;
  const float tiny = 1.17549435e-38f;
  float val = fmaxf(tiny, u * (1.0f - tiny) + tiny);
  float g   = -logf(-logf(val));

  bool  valid = (dpad[t * BZ + lane] != 1);
  float best  = valid ? g : -__builtin_inff();
  int   bidx  = lane;
  #pragma unroll
  for (int off = 16; off > 0; off >>= 1) {
    float ob = __shfl_xor(best, off, 32);
    int   oi = __shfl_xor(bidx, off, 32);
    if (ob > best || (ob == best && oi < bidx)) { best = ob; bidx = oi; }
  }
  if (lane == 0) Lws[t] = bidx;
}

// ---------------------------------------------------------------------------
// K2: squared norms of the 2048 gathered rows via V_WMMA_F32_16X16X4_F32.
// One wave handles 16 rows: accumulate Gram = X * X^T over K=1024 in chunks
// of 4; the diagonal is ||row||^2. For this shape the A (16x4) and B (4x16)
// lane layouts of X and X^T coincide, so a single float2 load feeds both.
// ---------------------------------------------------------------------------
__global__ __launch_bounds__(32) void norm_kernel(const float* __restrict__ emb,
                                                  const int* __restrict__ Lws,
                                                  float* __restrict__ norm2ws) {
  const int lane = threadIdx.x;                 // 0..31
  const int tile = blockIdx.x;                  // 0..127 (16 rows each)
  const int r    = tile * 16 + (lane & 15);     // timestep handled by this lane
  const int Lr   = Lws[r];
  // A layout 16x4 f32: lanes 0-15 hold (M=lane, K=k0,k0+1); lanes 16-31 hold
  // (M=lane-16, K=k0+2,k0+3)  ->  per-lane base pointer + conditional +2.
  const float* rowp = emb + ((size_t)r * BZ + (size_t)Lr) * EMB + ((lane >= 16) ? 2 : 0);

  v8f acc = {};
  for (int k0 = 0; k0 < EMB; k0 += 4) {
    v2f a = *(const v2f*)(rowp + k0);           // 8B aligned global_load_b64
    acc = __builtin_amdgcn_wmma_f32_16x16x4_f32(
        /*neg_a=*/false, a, /*neg_b=*/false, a,
        /*c_mod=*/(short)0, acc, /*reuse_a=*/false, /*reuse_b=*/false);
  }

  // Diagonal of C/D (16x16 f32): lane m (m<8) holds D[m][m] in acc[m];
  // lane m+24 holds D[m+8][m+8] in acc[m] (lanes 16-31: M=j+8, N=lane-16).
  int d = (lane < 8) ? lane : (lane - 24);
  float n2 = 0.0f;
  #pragma unroll
  for (int j = 0; j < 8; ++j) if (d == j) n2 = acc[j];
  if (lane < 8)        norm2ws[tile * 16 + lane]        = n2;
  else if (lane >= 24) norm2ws[tile * 16 + (lane - 16)] = n2;
}

// ---------------------------------------------------------------------------
// K3: bulk streaming copy emb -> out. 256 MB > 192 MB L2, so use
// non-temporal b128 loads/stores to stream past the cache hierarchy.
// ---------------------------------------------------------------------------
__global__ __launch_bounds__(256) void copy_kernel(const v4f* __restrict__ src,
                                                   v4f* __restrict__ dst,
                                                   long n4) {
  long i      = (long)blockIdx.x * blockDim.x + threadIdx.x;
  long stride = (long)gridDim.x * blockDim.x;
  for (; i < n4; i += stride) {
    v4f v = __builtin_nontemporal_load(&src[i]);
    __builtin_nontemporal_store(v, &dst[i]);
  }
}

// ---------------------------------------------------------------------------
// K4: overwrite the selected row of each timestep with eword*(1+eps/||eword||)
// and append L (as float) after the tensor.
// ---------------------------------------------------------------------------
__global__ __launch_bounds__(256) void perturb_kernel(const float* __restrict__ emb,
                                                      const int* __restrict__ Lws,
                                                      const float* __restrict__ norm2ws,
                                                      float* __restrict__ out,
                                                      float* __restrict__ Lout) {
  const int t  = blockIdx.x;
  const int Lr = Lws[t];
  const float scale = 1.0f + EPSILON / sqrtf(norm2ws[t]);
  const size_t base = ((size_t)t * BZ + (size_t)Lr) * EMB;
  for (int k = threadIdx.x; k < EMB; k += 256)
    out[base + k] = emb[base + k] * scale;
  if (threadIdx.x == 0) Lout[t] = (float)Lr;
}

// ---------------------------------------------------------------------------
extern "C" void kernel_launch(void* const* d_in, const int* in_sizes, int n_in,
                              void* d_out, int out_size, void* d_ws, size_t ws_size,
                              hipStream_t stream) {
  const float* emb   = (const float*)d_in[0];      // (2048, 32, 1024) f32
  // d_in[1] = data (int64) is mathematically unused by the reference.
  const int*   dpad  = (const int*)d_in[2];        // (2048, 32) int32

  int*   Lws     = (int*)d_ws;                              // 2048 ints
  float* norm2ws = (float*)((char*)d_ws + TLEN * sizeof(int)); // 2048 floats

  float* out  = (float*)d_out;                     // A, then L
  float* Lout = out + (size_t)TLEN * BZ * EMB;

  // K1: Gumbel-argmax selection (one wave32 per timestep)
  select_kernel<<<dim3(TLEN), dim3(32), 0, stream>>>(dpad, Lws);

  // K2: row norms via f32 WMMA Gram diagonal (16 rows per wave)
  norm_kernel<<<dim3(TLEN / 16), dim3(32), 0, stream>>>(emb, Lws, norm2ws);

  // K3: streaming NT copy of the full tensor (bandwidth-bound path)
  long n4 = (long)in_sizes[0] / 4;                 // 16,777,216 float4s
  copy_kernel<<<dim3(8192), dim3(256), 0, stream>>>((const v4f*)emb, (v4f*)out, n4);

  // K4: scale the selected rows in-place in out; emit L
  perturb_kernel<<<dim3(TLEN), dim3(256), 0, stream>>>(emb, Lws, norm2ws, out, Lout);
}